// DemosaicMalvar_50002009260332
// MI455X (gfx1250) — compile-verified
//
#include <hip/hip_runtime.h>

// Demosaic (Malvar) for x:(B,4,512,512)f32 -> out:(B,3,1024,1024)f32, channels [b,g,r].
// Memory-bound stencil: stage Bayer tile to LDS via CDNA5 async global->LDS DMA
// (ASYNCcnt path), compute 5x5 sparse stencils from LDS, NT-store float2 pairs.

#define TQW 64         // quads per tile in x  (128 output px)
#define TQH 8          // quads per tile in y  (16 output px)
#define LROWS 20       // 2*TQH + 4 halo rows (in xf pixels)
#define LCOLS 66       // TQW + 2 halo cols (in w units, per parity plane)
#define LSTRIDE 68     // padded row stride (floats)

typedef __attribute__((ext_vector_type(2))) float f32x2;

__global__ __launch_bounds__(256)
void demosaic_malvar_kernel(const float* __restrict__ in, float* __restrict__ out) {
    // lds[0] = even-x columns of xf (channel 2*(y&1)), lds[1] = odd-x (channel 2*(y&1)+1)
    __shared__ float lds[2][LROWS][LSTRIDE];

    const int tid = threadIdx.x;
    const int b   = blockIdx.z;
    const int qi0 = blockIdx.y * TQH;   // quad-row base
    const int qj0 = blockIdx.x * TQW;   // quad-col base

    const int ybase = 2 * qi0 - 2;      // xf row of lds row 0
    const int wbase = qj0 - 1;          // w of lds col 0 (xf col = 2*w + parity)

    // ---- Stage tile into LDS with async global->LDS (CDNA5, ASYNCcnt) ----
    // 40 segments (20 rows x 2 parities) of 66 contiguous floats each.
    for (int e = tid; e < LROWS * 2 * LCOLS; e += 256) {
        const int seg = e / LCOLS;
        const int k   = e - seg * LCOLS;
        const int par = seg & 1;
        const int row = seg >> 1;
        const int y = ybase + row;
        const int w = wbase + k;
        float* lp = &lds[par][row][k];
        if ((unsigned)y < 1024u && (unsigned)w < 512u) {
            const int c = 2 * (y & 1) + par;
            const float* gp = in + ((((b << 2) + c) << 9) + (y >> 1)) * 512 + w;
            // low 32 bits of the generic pointer to a __shared__ object == LDS byte offset
            const unsigned lo = (unsigned)(unsigned long long)(void*)lp;
            asm volatile("global_load_async_to_lds_b32 %0, %1, off"
                         :: "v"(lo), "v"(gp) : "memory");
        } else {
            *lp = 0.0f;   // zero padding outside image (disjoint from async targets)
        }
    }
    asm volatile("s_wait_asynccnt 0" ::: "memory");
    __syncthreads();

    // ---- Compute: each thread does 2 quads (qy and qy+4) at qx ----
    const int qx  = tid & (TQW - 1);
    const int qyB = tid >> 6;           // 0..3
#pragma unroll
    for (int qq = 0; qq < 2; ++qq) {
        const int qy = qyB + 4 * qq;
        const int r0 = 2 * qy + 2;      // lds row of xf row 2*qi
        const int r1 = r0 + 1;
        const int ke = qx, k0 = qx + 1, k1 = qx + 2;

        // E-plane taps (even xf columns)
        const float e_m2_0 = lds[0][r0-2][k0];
        const float e_m1_0 = lds[0][r0-1][k0];
        const float e_m1_1 = lds[0][r0-1][k1];
        const float e_0_e  = lds[0][r0  ][ke];
        const float e_0_0  = lds[0][r0  ][k0];
        const float e_0_1  = lds[0][r0  ][k1];
        const float e_1_e  = lds[0][r1  ][ke];
        const float e_1_0  = lds[0][r1  ][k0];
        const float e_1_1  = lds[0][r1  ][k1];
        const float e_2_0  = lds[0][r0+2][k0];
        const float e_2_1  = lds[0][r0+2][k1];
        const float e_3_0  = lds[0][r0+3][k0];
        // O-plane taps (odd xf columns)
        const float o_m2_0 = lds[1][r0-2][k0];
        const float o_m1_e = lds[1][r0-1][ke];
        const float o_m1_0 = lds[1][r0-1][k0];
        const float o_0_e  = lds[1][r0  ][ke];
        const float o_0_0  = lds[1][r0  ][k0];
        const float o_0_1  = lds[1][r0  ][k1];
        const float o_1_e  = lds[1][r1  ][ke];
        const float o_1_0  = lds[1][r1  ][k0];
        const float o_1_1  = lds[1][r1  ][k1];
        const float o_2_e  = lds[1][r0+2][ke];
        const float o_2_0  = lds[1][r0+2][k0];
        const float o_3_0  = lds[1][r0+3][k0];

        // Shared subexpressions across the 8 stencils
        const float f2A   = e_m2_0 + e_2_0 + e_0_e + e_0_1;   // axial +/-2 around A(2i,2j)
        const float diagA = o_m1_e + o_m1_0 + o_1_e + o_1_0;
        const float diagB = e_m1_0 + e_m1_1 + e_1_0 + e_1_1;  // around B(2i,2j+1)
        const float v2B   = o_m2_0 + o_2_0;
        const float h2B   = o_0_e  + o_0_1;
        const float diagC = o_0_e + o_0_0 + o_2_e + o_2_0;    // around C(2i+1,2j)
        const float v2C   = e_m1_0 + e_3_0;
        const float h2C   = e_1_e + e_1_1;
        const float f2D   = o_m1_0 + o_3_0 + o_1_e + o_1_1;   // around D(2i+1,2j+1)
        const float diagD = e_0_0 + e_0_1 + e_2_0 + e_2_1;

        // Phase (0,0): raw R site -> G and B interpolated, R = xf(2i,2j+1)
        const float gA = 0.50f*e_0_0 + 0.25f*(e_m1_0 + e_1_0 + o_0_e + o_0_0) - 0.125f*f2A;
        const float bA = 0.75f*e_0_0 + 0.25f*diagA - 0.1875f*f2A;
        // Phase (0,1): R = R_AT_G1, B = R_AT_G2, G = xf(2i,2j)
        const float rB = 0.625f*o_0_0 + 0.5f*(e_0_0 + e_0_1) - 0.125f*diagB - 0.125f*h2B + 0.0625f*v2B;
        const float bB = 0.625f*o_0_0 + 0.5f*(o_m1_0 + o_1_0) - 0.125f*diagB - 0.125f*v2B + 0.0625f*h2B;
        // Phase (1,0): R = R_AT_G2, B = R_AT_G1, G = xf(2i+1,2j+1)
        const float rC = 0.625f*e_1_0 + 0.5f*(e_0_0 + e_2_0) - 0.125f*diagC - 0.125f*v2C + 0.0625f*h2C;
        const float bC = 0.625f*e_1_0 + 0.5f*(o_1_e + o_1_0) - 0.125f*diagC - 0.125f*h2C + 0.0625f*v2C;
        // Phase (1,1): G and R interpolated, B = xf(2i+1,2j)
        const float gD = 0.50f*o_1_0 + 0.25f*(o_0_0 + o_2_0 + e_1_0 + e_1_1) - 0.125f*f2D;
        const float rD = 0.75f*o_1_0 + 0.25f*diagD - 0.1875f*f2D;

        const int qi = qi0 + qy;
        const int qj = qj0 + qx;
        const long long Y0 = 2 * qi;
        const long long X0 = 2 * qj;
        float* ob  = out + (((long long)b * 3 + 0) * 1024 + Y0) * 1024 + X0;
        float* og  = out + (((long long)b * 3 + 1) * 1024 + Y0) * 1024 + X0;
        float* orr = out + (((long long)b * 3 + 2) * 1024 + Y0) * 1024 + X0;

        f32x2 v;
        v.x = bA;    v.y = bB;    __builtin_nontemporal_store(v, (f32x2*)ob);
        v.x = bC;    v.y = e_1_0; __builtin_nontemporal_store(v, (f32x2*)(ob + 1024));
        v.x = gA;    v.y = e_0_0; __builtin_nontemporal_store(v, (f32x2*)og);
        v.x = o_1_0; v.y = gD;    __builtin_nontemporal_store(v, (f32x2*)(og + 1024));
        v.x = o_0_0; v.y = rB;    __builtin_nontemporal_store(v, (f32x2*)orr);
        v.x = rC;    v.y = rD;    __builtin_nontemporal_store(v, (f32x2*)(orr + 1024));
    }
}

extern "C" void kernel_launch(void* const* d_in, const int* in_sizes, int n_in,
                              void* d_out, int out_size, void* d_ws, size_t ws_size,
                              hipStream_t stream) {
    (void)n_in; (void)out_size; (void)d_ws; (void)ws_size;
    const float* x = (const float*)d_in[0];
    float* out = (float*)d_out;
    const int B = in_sizes[0] / (4 * 512 * 512);   // 8
    dim3 grid(512 / TQW, 512 / TQH, B);            // (8, 64, B)
    demosaic_malvar_kernel<<<grid, dim3(256), 0, stream>>>(x, out);
}